// SSIM_4277787427036
// MI455X (gfx1250) — compile-verified
//
#include <hip/hip_runtime.h>
#include <math.h>

typedef __attribute__((ext_vector_type(2))) float v2f;
typedef __attribute__((ext_vector_type(8))) float v8f;

#define WIN    11
#define RAD    5
#define TS     32          // output region per workgroup (32x32)
#define RH     42          // region + halo (32 + 10)
#define RW     42
#define XPITCH 44          // LDS pitch for staged inputs
#define IPITCH 48          // LDS pitch for intermediates (conflict-free half-wave row split)
#define HN     512
#define WN     512
#define NPLANE 48          // 16*3
#define C1F    0.0001f     // 0.01^2
#define C2F    0.0009f     // 0.03^2
#define EPSF   1e-8f

// Vertical 11-tap conv of one 16x16 tile as W(16x28,banded) x I(28x16),
// chained through 7 x v_wmma_f32_16x16x4_f32.
__device__ __forceinline__ v8f vconv_wmma(const float (&I)[RH][IPITCH],
                                          int T, int Cc, int col, int hi,
                                          const v2f* aw) {
  v8f acc = {};
#pragma unroll
  for (int c = 0; c < 7; ++c) {
    int r0 = T + 4 * c + hi;           // row for this lane-half (K = 4c+hi)
    if (r0 > RH - 1) r0 = RH - 1;      // clamped rows always carry zero weight
    int r1 = r0 + 1;
    if (r1 > RH - 1) r1 = RH - 1;
    v2f b;
    b.x = I[r0][Cc + col];
    b.y = I[r1][Cc + col];
    acc = __builtin_amdgcn_wmma_f32_16x16x4_f32(
        /*neg_a=*/false, aw[c], /*neg_b=*/false, b,
        /*c_mod=*/(short)0, acc, /*reuse_a=*/false, /*reuse_b=*/false);
  }
  return acc;
}

__global__ __launch_bounds__(128) void ssim_kernel(const float* __restrict__ x,
                                                   const float* __restrict__ y,
                                                   double* __restrict__ gsum) {
  __shared__ float smx[RH][XPITCH];
  __shared__ float smy[RH][XPITCH];
  __shared__ float smi[5][RH][IPITCH];   // conv_h of x, y, x^2, y^2, xy
  __shared__ float wz[48];               // banded-weight lookup: wz[d+15] = g[d] (0<=d<=10) else 0
  __shared__ float wsum[4];

  const int tid = threadIdx.x;

  // Gaussian window, sigma = 1.5, normalized (registers only; literal indices).
  float g[WIN];
  float gs = 0.f;
#pragma unroll
  for (int i = 0; i < WIN; ++i) {
    float d = (float)(i - RAD);
    float v = __expf(-(d * d) * (1.0f / 4.5f));
    g[i] = v;
    gs += v;
  }
  const float ginv = 1.f / gs;
#pragma unroll
  for (int i = 0; i < WIN; ++i) g[i] *= ginv;

  if (tid < 48) {
    int d = tid - 15;
    float v = 0.f;
    if (d >= 0 && d <= 10) {
      float dd = (float)(d - RAD);
      v = __expf(-(dd * dd) * (1.0f / 4.5f)) * ginv;
    }
    wz[tid] = v;
  }

  // Which 32x32 region of which (b,c) plane.
  const int bid   = blockIdx.x;
  const int plane = bid >> 8;            // 256 regions per 512x512 plane
  const int rem   = bid & 255;
  const int R0    = (rem >> 4) << 5;
  const int C0    = (rem & 15) << 5;
  const float* __restrict__ xp = x + (size_t)plane * (HN * WN);
  const float* __restrict__ yp = y + (size_t)plane * (HN * WN);

  // Stage 42x42 halo of x and y into LDS (zero padding == SAME).
  for (int i = tid; i < RH * RW; i += 128) {
    int r = i / RW, c = i - r * RW;
    int gr = R0 - RAD + r;
    int gc = C0 - RAD + c;
    bool ok = (gr >= 0) & (gr < HN) & (gc >= 0) & (gc < WN);
    size_t off = (size_t)gr * WN + gc;
    float xv = ok ? xp[off] : 0.f;
    float yv = ok ? yp[off] : 0.f;
    smx[r][c] = xv;
    smy[r][c] = yv;
  }
  __syncthreads();

  // Horizontal 11-tap pass for all 5 statistics (VALU FMAs out of LDS).
  for (int i = tid; i < RH * TS; i += 128) {
    int p = i >> 5;
    int c = i & 31;
    float sx = 0.f, sy = 0.f, sxx = 0.f, syy = 0.f, sxy = 0.f;
#pragma unroll
    for (int t = 0; t < WIN; ++t) {
      float w  = g[t];
      float xv = smx[p][c + t];
      float yv = smy[p][c + t];
      sx  = fmaf(w, xv, sx);
      sy  = fmaf(w, yv, sy);
      sxx = fmaf(w * xv, xv, sxx);
      syy = fmaf(w * yv, yv, syy);
      sxy = fmaf(w * xv, yv, sxy);
    }
    smi[0][p][c] = sx;
    smi[1][p][c] = sy;
    smi[2][p][c] = sxx;
    smi[3][p][c] = syy;
    smi[4][p][c] = sxy;
  }
  __syncthreads();

  // Vertical pass: each wave owns one 16x16 tile of the 32x32 region.
  const int wave = tid >> 5;
  const int lane = tid & 31;
  const int T    = (wave >> 1) << 4;     // tile row offset in region
  const int Cc   = (wave & 1) << 4;      // tile col offset in region
  const int m    = lane & 15;            // A-matrix row / B-matrix column (ISA 16x4 f32 layout)
  const int hi   = (lane >> 4) << 1;     // lanes 0-15 -> K+0/1, lanes 16-31 -> K+2/3

  // A-matrix chunks (banded Gaussian), shared across all 5 statistics.
  v2f aw[7];
#pragma unroll
  for (int c = 0; c < 7; ++c) {
    int k0 = 4 * c + hi;
    aw[c].x = wz[k0 - m + 15];
    aw[c].y = wz[k0 + 1 - m + 15];
  }

  v8f aX  = vconv_wmma(smi[0], T, Cc, m, hi, aw);
  v8f aY  = vconv_wmma(smi[1], T, Cc, m, hi, aw);
  v8f aXX = vconv_wmma(smi[2], T, Cc, m, hi, aw);
  v8f aYY = vconv_wmma(smi[3], T, Cc, m, hi, aw);
  v8f aXY = vconv_wmma(smi[4], T, Cc, m, hi, aw);

  // Fused SSIM map + clamp; each lane holds 8 pixels of the tile.
  float lsum = 0.f;
#pragma unroll
  for (int i = 0; i < 8; ++i) {
    float mx = aX[i], my = aY[i];
    float mx2 = mx * mx, my2 = my * my, mxy = mx * my;
    float sxv  = aXX[i] - mx2;
    float syv  = aYY[i] - my2;
    float sxyv = aXY[i] - mxy;
    float n = (2.f * mxy + C1F) * (2.f * sxyv + C2F);
    float d = (mx2 + my2 + C1F) * (sxv + syv + C2F);
    float s = n / (d + EPSF);
    s = 0.5f * (1.f + s);
    s = fminf(fmaxf(s, 0.f), 1.f);
    lsum += s;
  }

  // Wave32 reduce -> block reduce -> one fp64 atomic per workgroup.
#pragma unroll
  for (int o = 16; o > 0; o >>= 1) lsum += __shfl_xor(lsum, o, 32);
  if (lane == 0) wsum[wave] = lsum;
  __syncthreads();
  if (tid == 0) {
    double bs = (double)wsum[0] + (double)wsum[1] + (double)wsum[2] + (double)wsum[3];
    atomicAdd(gsum, bs);
  }
}

__global__ void zero_kernel(double* p) { p[0] = 0.0; }

__global__ void final_kernel(const double* __restrict__ p, float* __restrict__ out) {
  out[0] = (float)(p[0] * (1.0 / (16.0 * 3.0 * 512.0 * 512.0)));
}

extern "C" void kernel_launch(void* const* d_in, const int* in_sizes, int n_in,
                              void* d_out, int out_size, void* d_ws, size_t ws_size,
                              hipStream_t stream) {
  const float* x = (const float*)d_in[0];
  const float* y = (const float*)d_in[1];
  double* acc = (double*)d_ws;
  float* out  = (float*)d_out;

  zero_kernel<<<1, 1, 0, stream>>>(acc);
  ssim_kernel<<<NPLANE * 256, 128, 0, stream>>>(x, y, acc);
  final_kernel<<<1, 1, 0, stream>>>(acc, out);
}